// StackedGNN_35150012351303
// MI455X (gfx1250) — compile-verified
//
#include <hip/hip_runtime.h>
#include <math.h>

typedef __attribute__((ext_vector_type(2))) float v2f;
typedef __attribute__((ext_vector_type(8))) float v8f;

#define FIELD_DIM 64
#define N_FIELDS  20

// ---------------------------------------------------------------------------
// zero fill
// ---------------------------------------------------------------------------
__global__ void zero_f32(float* __restrict__ p, long n) {
    long i = (long)blockIdx.x * blockDim.x + threadIdx.x;
    if (i < n) p[i] = 0.0f;
}

// ---------------------------------------------------------------------------
// user[n][d] = mean_f emb[field_index[n][f]][d]     (one 64-thread block/node)
// ---------------------------------------------------------------------------
__global__ void embed_mean(const int* __restrict__ fidx,
                           const float* __restrict__ emb,
                           float* __restrict__ user, int n_nodes) {
    __shared__ int sidx[N_FIELDS];
    int node = blockIdx.x;
    if (node >= n_nodes) return;
    if (threadIdx.x < N_FIELDS) sidx[threadIdx.x] = fidx[node * N_FIELDS + threadIdx.x];
    __syncthreads();
    int d = threadIdx.x;  // blockDim == 64
    float s = 0.0f;
    #pragma unroll
    for (int f = 0; f < N_FIELDS; ++f)
        s += emb[(size_t)sidx[f] * FIELD_DIM + d];
    user[(size_t)node * FIELD_DIM + d] = s * (1.0f / N_FIELDS);
}

// ---------------------------------------------------------------------------
// deg[dst[e]] += 1
// ---------------------------------------------------------------------------
__global__ void degree_count(const int* __restrict__ dst, float* __restrict__ deg,
                             int n_edges) {
    int e = blockIdx.x * blockDim.x + threadIdx.x;
    if (e < n_edges) atomicAdd(&deg[dst[e]], 1.0f);
}

// ---------------------------------------------------------------------------
// accum[dst[e]][:] += feat[src[e]][:]   (16 threads/edge, float4 each)
// ---------------------------------------------------------------------------
__global__ void scatter_add(const float* __restrict__ feat,
                            const int* __restrict__ src,
                            const int* __restrict__ dst,
                            float* __restrict__ accum, int n_edges) {
    unsigned gid = blockIdx.x * blockDim.x + threadIdx.x;
    int e = gid >> 4;
    if (e >= n_edges) return;
    int q = (gid & 15) * 4;
    int s = src[e], d = dst[e];
    float4 v = *reinterpret_cast<const float4*>(feat + (size_t)s * FIELD_DIM + q);
    float* a = accum + (size_t)d * FIELD_DIM + q;
    atomicAdd(a + 0, v.x);
    atomicAdd(a + 1, v.y);
    atomicAdd(a + 2, v.z);
    atomicAdd(a + 3, v.w);
}

// ---------------------------------------------------------------------------
// out[r][o] = act( (accum[r][:]/max(deg[r],1)) . W[o][:] + bias[o] )
// W is [out_ch, 64] row-major. One wave -> 16 rows x 16 cols per WMMA tile.
// V_WMMA_F32_16X16X4_F32, 16 k-steps of K=4 covering K=64.
// ---------------------------------------------------------------------------
__global__ void mlp_wmma(const float* __restrict__ accum,
                         const float* __restrict__ deg,
                         const float* __restrict__ W,
                         const float* __restrict__ bias,
                         float* __restrict__ out,
                         int n_rows, int out_ch, int out_stride, int do_relu) {
    __shared__ float sW[64 * 64];  // weights zero-padded to 64 output rows
    int tid = threadIdx.x;
    for (int i = tid; i < 64 * 64; i += blockDim.x) {
        int o = i >> 6, k = i & 63;
        sW[i] = (o < out_ch) ? W[o * 64 + k] : 0.0f;
    }
    __syncthreads();

    int wave = tid >> 5;
    int rowtile = blockIdx.x * (blockDim.x >> 5) + wave;
    int rowbase = rowtile * 16;
    if (rowbase >= n_rows) return;          // wave-uniform: EXEC stays all-1s

    int lane = tid & 31;
    int half = lane >> 4;                    // 0: K={0,1}, 1: K={2,3}
    int lm   = lane & 15;                    // A: M row / B: N col

    int arow = rowbase + lm;
    float inv = 1.0f / fmaxf(deg[arow], 1.0f);
    const float* ap = accum + (size_t)arow * 64;

    // Preload all A fragments (reused across every output-column tile)
    v2f a[16];
    #pragma unroll
    for (int ks = 0; ks < 16; ++ks) {
        int k = ks * 4 + half * 2;
        a[ks].x = ap[k + 0] * inv;
        a[ks].y = ap[k + 1] * inv;
    }

    int n_tiles = (out_ch + 15) >> 4;
    for (int t = 0; t < n_tiles; ++t) {
        int o0 = t * 16;
        v8f c = {};
        #pragma unroll
        for (int ks = 0; ks < 16; ++ks) {
            int k = ks * 4 + half * 2;
            v2f b;
            b.x = sW[(o0 + lm) * 64 + k + 0];
            b.y = sW[(o0 + lm) * 64 + k + 1];
            c = __builtin_amdgcn_wmma_f32_16x16x4_f32(
                    /*neg_a=*/false, a[ks], /*neg_b=*/false, b,
                    /*c_mod=*/(short)0, c, /*reuse_a=*/false, /*reuse_b=*/false);
        }
        int o = o0 + lm;
        if (o < out_ch) {
            float bv = bias[o];
            #pragma unroll
            for (int j = 0; j < 8; ++j) {
                int r = rowbase + j + half * 8;
                float v = c[j] + bv;
                if (do_relu) v = fmaxf(v, 0.0f);
                out[(size_t)r * out_stride + o] = v;
            }
        }
    }
}

// ---------------------------------------------------------------------------
// in-place log_softmax over 40 channels, one thread per row
// ---------------------------------------------------------------------------
__global__ void logsoftmax40(float* __restrict__ out, int n_rows) {
    int r = blockIdx.x * blockDim.x + threadIdx.x;
    if (r >= n_rows) return;
    float* p = out + (size_t)r * 40;
    float v[40];
    float m = -INFINITY;
    #pragma unroll
    for (int i = 0; i < 40; ++i) { v[i] = p[i]; m = fmaxf(m, v[i]); }
    float s = 0.0f;
    #pragma unroll
    for (int i = 0; i < 40; ++i) s += __expf(v[i] - m);
    float ls = __logf(s);
    #pragma unroll
    for (int i = 0; i < 40; ++i) p[i] = v[i] - m - ls;
}

// ---------------------------------------------------------------------------
extern "C" void kernel_launch(void* const* d_in, const int* in_sizes, int n_in,
                              void* d_out, int out_size, void* d_ws, size_t ws_size,
                              hipStream_t stream) {
    const int*   fidx = (const int*)d_in[0];   // [N, 20]
    const int*   eidx = (const int*)d_in[1];   // [2, E]
    const float* emb  = (const float*)d_in[2]; // [2000, 64]
    const float* W1   = (const float*)d_in[3]; // [64, 64]
    const float* b1   = (const float*)d_in[4]; // [64]
    const float* W2   = (const float*)d_in[5]; // [40, 64]
    const float* b2   = (const float*)d_in[6]; // [40]

    const int N = in_sizes[0] / N_FIELDS;      // 100000
    const int E = in_sizes[1] / 2;             // 3200000
    const int OUT_CH = out_size / N;           // 40

    const int* src = eidx;
    const int* dst = eidx + E;

    float* feat0 = (float*)d_ws;               // user, later h   [N*64]
    float* accum = feat0 + (size_t)N * 64;     //                 [N*64]
    float* deg   = accum + (size_t)N * 64;     //                 [N]

    float* outp = (float*)d_out;

    long nfeat = (long)N * 64;
    dim3 b256(256);

    // pass 0: clear accum + deg, build user features + degree
    zero_f32<<<dim3((unsigned)((nfeat + 255) / 256)), b256, 0, stream>>>(accum, nfeat);
    zero_f32<<<dim3((unsigned)((N + 255) / 256)), b256, 0, stream>>>(deg, N);
    embed_mean<<<dim3((unsigned)N), dim3(64), 0, stream>>>(fidx, emb, feat0, N);
    degree_count<<<dim3((unsigned)((E + 255) / 256)), b256, 0, stream>>>(dst, deg, E);

    // layer 1: mean-agg(user) -> relu(. @ W1^T + b1) = h (reuses feat0)
    unsigned sblocks = (unsigned)(((long)E * 16 + 255) / 256);
    scatter_add<<<dim3(sblocks), b256, 0, stream>>>(feat0, src, dst, accum, E);
    int rowtiles = (N + 15) / 16;
    unsigned mblocks = (unsigned)((rowtiles + 7) / 8);
    mlp_wmma<<<dim3(mblocks), b256, 0, stream>>>(accum, deg, W1, b1, feat0,
                                                 N, 64, 64, /*relu=*/1);

    // layer 2: mean-agg(h) -> . @ W2^T + b2 straight into d_out
    zero_f32<<<dim3((unsigned)((nfeat + 255) / 256)), b256, 0, stream>>>(accum, nfeat);
    scatter_add<<<dim3(sblocks), b256, 0, stream>>>(feat0, src, dst, accum, E);
    mlp_wmma<<<dim3(mblocks), b256, 0, stream>>>(accum, deg, W2, b2, outp,
                                                 N, OUT_CH, OUT_CH, /*relu=*/0);

    // log_softmax in place on d_out
    logsoftmax40<<<dim3((unsigned)((N + 255) / 256)), b256, 0, stream>>>(outp, N);
}